// GNNSurrogate_11295763988987
// MI455X (gfx1250) — compile-verified
//
#include <hip/hip_runtime.h>

#define IN_DIM 128
#define HID    64
#define OUTD   2

typedef __attribute__((ext_vector_type(2))) float v2f;
typedef __attribute__((ext_vector_type(8))) float v8f;

__device__ __forceinline__ void atomAddF(float* p, float v) {
    __hip_atomic_fetch_add(p, v, __ATOMIC_RELAXED, __HIP_MEMORY_SCOPE_AGENT);
}

// deg[i] = 1.0 (self loop)
__global__ void k_init_deg(float* __restrict__ deg, int n) {
    int i = blockIdx.x * blockDim.x + threadIdx.x;
    if (i < n) deg[i] = 1.0f;
}

// deg[dst] += 1 for each real edge
__global__ void k_count(const int* __restrict__ ei, float* __restrict__ deg, int E) {
    int e = blockIdx.x * blockDim.x + threadIdx.x;
    if (e < E) atomAddF(&deg[ei[E + e]], 1.0f);
}

// dinv = rsqrt(deg)  (deg >= 1 always due to self loop)
__global__ void k_dinv(float* __restrict__ deg, int n) {
    int i = blockIdx.x * blockDim.x + threadIdx.x;
    if (i < n) deg[i] = rsqrtf(deg[i]);
}

// h = x @ W1 via V_WMMA_F32_16X16X4_F32; also initialize the layer-1
// accumulator with bias + self-loop contribution: agg = b1 + dinv[i]^2 * h[i]
__global__ void k_gemm1(const float* __restrict__ x, const float* __restrict__ W1,
                        const float* __restrict__ b1, const float* __restrict__ dinv,
                        float* __restrict__ h, float* __restrict__ agg,
                        int n_nodes, int mtiles) {
    int wid  = blockIdx.x * (blockDim.x >> 5) + (threadIdx.x >> 5);
    int lane = threadIdx.x & 31;
    int tm = wid >> 2;      // HID/16 = 4 column tiles
    int tn = wid & 3;
    if (tm >= mtiles) return;          // wave-uniform: EXEC stays all-1s

    int row0 = tm * 16, col0 = tn * 16;
    int m    = lane & 15;
    int half = lane >> 4;              // 0 or 1
    int koff = half << 1;              // 0 or 2

    int arow = row0 + m;
    if (arow >= n_nodes) arow = n_nodes - 1;   // clamp (reads only)
    const float* xrow = x + (size_t)arow * IN_DIM;

    v8f c = {};
    #pragma unroll 4
    for (int k = 0; k < IN_DIM; k += 4) {
        // A 16x4 f32: lanes 0-15 -> K = k+0,k+1 ; lanes 16-31 -> K = k+2,k+3
        v2f a = *reinterpret_cast<const v2f*>(xrow + k + koff);   // 8B aligned
        // B 4x16 f32: mirrored K split across lane halves, N = lane&15
        v2f b;
        b.x = W1[(size_t)(k + koff)     * HID + col0 + m];
        b.y = W1[(size_t)(k + koff + 1) * HID + col0 + m];
        c = __builtin_amdgcn_wmma_f32_16x16x4_f32(
                false, a, false, b, (short)0, c, false, false);
    }

    // D 16x16 f32: VGPR g -> row (g + 8*half), col = lane&15
    int col   = col0 + m;
    float b1c = b1[col];
    int rbase = row0 + (half << 3);

    if (row0 + 16 <= n_nodes) {
        // full tile: straight-line stores, no per-row exec juggling
        #pragma unroll
        for (int g = 0; g < 8; ++g) {
            int row   = rbase + g;
            float di  = dinv[row];
            size_t idx = (size_t)row * HID + col;
            float v = c[g];
            h[idx]   = v;
            agg[idx] = b1c + di * di * v;   // self-loop: norm = dinv^2
        }
    } else {
        #pragma unroll
        for (int g = 0; g < 8; ++g) {
            int row = rbase + g;
            if (row < n_nodes) {
                float di  = dinv[row];
                size_t idx = (size_t)row * HID + col;
                float v = c[g];
                h[idx]   = v;
                agg[idx] = b1c + di * di * v;
            }
        }
    }
}

// Layer-1 message passing: 16 threads per edge, 4 dims each (float4 gather,
// 4 fp32 atomics). h is L2-resident (25.6 MB << 192 MB L2).
__global__ void k_scatter1(const int* __restrict__ ei, const float* __restrict__ dinv,
                           const float* __restrict__ h, float* __restrict__ agg,
                           int E) {
    long long tid = (long long)blockIdx.x * blockDim.x + threadIdx.x;
    int e = (int)(tid >> 4);
    if (e >= E) return;
    int t = (int)(tid & 15);
    int s = ei[e];
    int d = ei[E + e];
    float w = dinv[s] * dinv[d];
    const float4 hv = *reinterpret_cast<const float4*>(h + (size_t)s * HID + t * 4);
    float* ap = agg + (size_t)d * HID + t * 4;
    atomAddF(ap + 0, w * hv.x);
    atomAddF(ap + 1, w * hv.y);
    atomAddF(ap + 2, w * hv.z);
    atomAddF(ap + 3, w * hv.w);
}

// h2 = relu(agg) @ W2 ; out initialized with b2 + self-loop term
__global__ void k_gemm2(const float* __restrict__ agg, const float* __restrict__ W2,
                        const float* __restrict__ b2, const float* __restrict__ dinv,
                        float* __restrict__ h2, float* __restrict__ out, int n) {
    int i = blockIdx.x * blockDim.x + threadIdx.x;
    if (i >= n) return;
    const float4* a4 = reinterpret_cast<const float4*>(agg + (size_t)i * HID);
    float s0 = 0.0f, s1 = 0.0f;
    #pragma unroll
    for (int q = 0; q < HID / 4; ++q) {       // 16 x b128 loads
        float4 v = a4[q];
        int d = q * 4;
        float r0 = fmaxf(v.x, 0.0f), r1 = fmaxf(v.y, 0.0f);
        float r2 = fmaxf(v.z, 0.0f), r3 = fmaxf(v.w, 0.0f);
        s0 = fmaf(r0, W2[2*d + 0], s0); s1 = fmaf(r0, W2[2*d + 1], s1);
        s0 = fmaf(r1, W2[2*d + 2], s0); s1 = fmaf(r1, W2[2*d + 3], s1);
        s0 = fmaf(r2, W2[2*d + 4], s0); s1 = fmaf(r2, W2[2*d + 5], s1);
        s0 = fmaf(r3, W2[2*d + 6], s0); s1 = fmaf(r3, W2[2*d + 7], s1);
    }
    float2* h2v = reinterpret_cast<float2*>(h2 + 2 * (size_t)i);
    *h2v = make_float2(s0, s1);
    float w = dinv[i] * dinv[i];
    float2* ov = reinterpret_cast<float2*>(out + 2 * (size_t)i);
    *ov = make_float2(b2[0] + w * s0, b2[1] + w * s1);
}

// Layer-2 message passing: 1 thread per edge, 2 dims (float2 gather)
__global__ void k_scatter2(const int* __restrict__ ei, const float* __restrict__ dinv,
                           const float* __restrict__ h2, float* __restrict__ out,
                           int E) {
    int e = blockIdx.x * blockDim.x + threadIdx.x;
    if (e >= E) return;
    int s = ei[e];
    int d = ei[E + e];
    float w = dinv[s] * dinv[d];
    float2 hv = *reinterpret_cast<const float2*>(h2 + 2 * (size_t)s);
    atomAddF(&out[2 * d + 0], w * hv.x);
    atomAddF(&out[2 * d + 1], w * hv.y);
}

extern "C" void kernel_launch(void* const* d_in, const int* in_sizes, int n_in,
                              void* d_out, int out_size, void* d_ws, size_t ws_size,
                              hipStream_t stream) {
    const float* x  = (const float*)d_in[0];
    const int*   ei = (const int*)  d_in[1];   // [2, E] int32 (JAX x64 off)
    const float* W1 = (const float*)d_in[2];
    const float* b1 = (const float*)d_in[3];
    const float* W2 = (const float*)d_in[4];
    const float* b2 = (const float*)d_in[5];
    float* out = (float*)d_out;

    const int n_nodes = in_sizes[0] / IN_DIM;
    const int E       = in_sizes[1] / 2;

    // workspace carve-out (floats): dinv | h | agg | h2   (~52 MB total)
    float* ws   = (float*)d_ws;
    size_t off  = 0;
    float* dinv = ws + off; off += (size_t)((n_nodes + 127) / 128) * 128;
    float* h    = ws + off; off += (size_t)n_nodes * HID;
    float* agg  = ws + off; off += (size_t)n_nodes * HID;
    float* h2   = ws + off;

    const int B = 256;
    k_init_deg<<<(n_nodes + B - 1) / B, B, 0, stream>>>(dinv, n_nodes);
    k_count  <<<(E + B - 1) / B,       B, 0, stream>>>(ei, dinv, E);
    k_dinv   <<<(n_nodes + B - 1) / B, B, 0, stream>>>(dinv, n_nodes);

    const int mtiles = (n_nodes + 15) / 16;
    const int waves  = mtiles * 4;                 // 4 column tiles (HID=64)
    k_gemm1<<<(waves + 7) / 8, B, 0, stream>>>(x, W1, b1, dinv, h, agg,
                                               n_nodes, mtiles);

    long long s1threads = (long long)E * 16;
    k_scatter1<<<(unsigned int)((s1threads + B - 1) / B), B, 0, stream>>>(
        ei, dinv, h, agg, E);

    k_gemm2<<<(n_nodes + B - 1) / B, B, 0, stream>>>(agg, W2, b2, dinv, h2, out,
                                                     n_nodes);

    k_scatter2<<<(E + B - 1) / B, B, 0, stream>>>(ei, dinv, h2, out, E);
}